// GNN_Local_33251636806000
// MI455X (gfx1250) — compile-verified
//
#include <hip/hip_runtime.h>
#include <hip/hip_bf16.h>

#define DIM 32
#define NEG_SLOPE 0.01f
#define BN_EPS 1e-5f

typedef __attribute__((ext_vector_type(2))) float v2f;
typedef __attribute__((ext_vector_type(8))) float v8f;

// Hardware f32 atomic add (no CAS loop). L2-resident traffic dominates this
// workload, so we want native global_atomic_add_f32.
__device__ __forceinline__ void atomAddF(float* p, float v) {
#if __has_builtin(__builtin_amdgcn_global_atomic_fadd_f32)
    __builtin_amdgcn_global_atomic_fadd_f32(p, v);
#else
    unsafeAtomicAdd(p, v);
#endif
}

__global__ void zero_kernel(float* __restrict__ p, int n) {
    int i = blockIdx.x * blockDim.x + threadIdx.x;
    if (i < n) p[i] = 0.0f;
}

__global__ void deg_kernel(const long long* __restrict__ dst,
                           const float* __restrict__ w,
                           float* __restrict__ deg, int E) {
    int e = blockIdx.x * blockDim.x + threadIdx.x;
    if (e < E) atomAddF(&deg[(int)dst[e]], w[e]);
}

__global__ void norm_kernel(const long long* __restrict__ src,
                            const long long* __restrict__ dst,
                            const float* __restrict__ w,
                            const float* __restrict__ deg,
                            float* __restrict__ nrm, int E) {
    int e = blockIdx.x * blockDim.x + threadIdx.x;
    if (e >= E) return;
    float ds = deg[(int)src[e]];
    float dd = deg[(int)dst[e]];
    float a = ds > 0.0f ? rsqrtf(ds) : 0.0f;
    float b = dd > 0.0f ? rsqrtf(dd) : 0.0f;
    nrm[e] = a * w[e] * b;
}

// hout[dst,:] += norm[e] * hin[src,:]
// 8 lanes per edge, float4 per lane: each edge's 128B source row is one
// coalesced burst across its lane group; all rows are L2-resident.
__global__ void prop_kernel(const long long* __restrict__ src,
                            const long long* __restrict__ dst,
                            const float* __restrict__ nrm,
                            const float* __restrict__ hin,
                            float* __restrict__ hout, int E) {
    int t = blockIdx.x * blockDim.x + threadIdx.x;
    int e = t >> 3;
    if (e >= E) return;
    int c = (t & 7) * 4;
    int s = (int)src[e];
    int d = (int)dst[e];
    float w = nrm[e];
    const float4 h = *(const float4*)(hin + (size_t)s * DIM + c);
    float* o = hout + (size_t)d * DIM + c;
    atomAddF(o + 0, w * h.x);
    atomAddF(o + 1, w * h.y);
    atomAddF(o + 2, w * h.z);
    atomAddF(o + 3, w * h.w);
}

// acc[16x32 tile] (+)= h[16x32 tile] @ W[32x32]  via V_WMMA_F32_16X16X4_F32.
// One wave per 16-node tile; 2 col-tiles x 8 K-chunks = 16 WMMA per wave.
// accumulate==0: seed C with broadcast bias; accumulate==1: seed C from acc.
__global__ void __launch_bounds__(256)
tag_gemm_kernel(const float* __restrict__ h, const float* __restrict__ W,
                const float* __restrict__ bias, float* __restrict__ acc,
                int nTiles, int accumulate) {
    int wave = (blockIdx.x * blockDim.x + threadIdx.x) >> 5;
    if (wave >= nTiles) return;          // wave-uniform: EXEC all-1 for WMMA
    int lane = threadIdx.x & 31;
    int half = lane >> 4;                // 0: lanes 0-15, 1: lanes 16-31
    int l16  = lane & 15;
    int r0   = wave * 16;

    // C/D 16x16 f32 layout: VGPR j holds row (j + half*8), col = l16
    v8f c[2];
    if (accumulate) {
#pragma unroll
        for (int ct = 0; ct < 2; ++ct)
#pragma unroll
            for (int j = 0; j < 8; ++j)
                c[ct][j] = acc[(size_t)(r0 + j + half * 8) * DIM + ct * 16 + l16];
    } else {
#pragma unroll
        for (int ct = 0; ct < 2; ++ct) {
            float bv = bias[ct * 16 + l16];
#pragma unroll
            for (int j = 0; j < 8; ++j) c[ct][j] = bv;
        }
    }

#pragma unroll
    for (int kk = 0; kk < 8; ++kk) {
        // A 16x4 f32 layout: lane row = l16, VGPR{0,1} = K {kb, kb+1},
        // lanes 16-31 carry K+2 (kb = kk*4 + half*2, always even -> b64 load)
        int kb = kk * 4 + half * 2;
        v2f a = *(const v2f*)(h + (size_t)(r0 + l16) * DIM + kb);
#pragma unroll
        for (int ct = 0; ct < 2; ++ct) {
            v2f b;   // B 4x16: lane col = l16, same K striping as A
            b.x = W[(kb + 0) * DIM + ct * 16 + l16];
            b.y = W[(kb + 1) * DIM + ct * 16 + l16];
            c[ct] = __builtin_amdgcn_wmma_f32_16x16x4_f32(
                false, a, false, b, (short)0, c[ct], false, false);
        }
    }

#pragma unroll
    for (int ct = 0; ct < 2; ++ct)
#pragma unroll
        for (int j = 0; j < 8; ++j)
            acc[(size_t)(r0 + j + half * 8) * DIM + ct * 16 + l16] = c[ct][j];
}

// Per-node BatchNorm over the 32-feature dim + LeakyReLU.
// One wave32 per node: lane = feature; butterfly reductions via __shfl_xor.
__global__ void __launch_bounds__(256)
bn_lrelu_kernel(const float* __restrict__ in, const float* __restrict__ gamma,
                const float* __restrict__ beta, float* __restrict__ out, int N) {
    int t = blockIdx.x * blockDim.x + threadIdx.x;
    int node = t >> 5;
    if (node >= N) return;
    int lane = t & 31;
    float v = in[(size_t)node * DIM + lane];
    float s = v;
#pragma unroll
    for (int off = 16; off > 0; off >>= 1) s += __shfl_xor(s, off);
    float mu = s * (1.0f / DIM);
    float d = v - mu;
    float q = d * d;
#pragma unroll
    for (int off = 16; off > 0; off >>= 1) q += __shfl_xor(q, off);
    float var = q * (1.0f / DIM);
    float y = d * rsqrtf(var + BN_EPS) * gamma[node] + beta[node];
    out[(size_t)node * DIM + lane] = y > 0.0f ? y : NEG_SLOPE * y;
}

__global__ void mask_kernel(const float* __restrict__ in,
                            const float* __restrict__ mask,
                            float* __restrict__ out, int total) {
    int i = blockIdx.x * blockDim.x + threadIdx.x;
    if (i < total) out[i] = in[i] * mask[i >> 5];  // DIM == 32
}

extern "C" void kernel_launch(void* const* d_in, const int* in_sizes, int n_in,
                              void* d_out, int out_size, void* d_ws, size_t ws_size,
                              hipStream_t stream) {
    const float*     x     = (const float*)d_in[0];
    const long long* ei    = (const long long*)d_in[1];   // int64 [2,E]
    const float*     ew    = (const float*)d_in[2];
    const float*     fmask = (const float*)d_in[3];
    const float*     W0    = (const float*)d_in[4];       // [4,32,32]
    const float*     b0    = (const float*)d_in[5];
    const float*     W1    = (const float*)d_in[6];
    const float*     b1    = (const float*)d_in[7];
    const float*     gamma = (const float*)d_in[8];
    const float*     beta  = (const float*)d_in[9];

    const int N = in_sizes[0] / DIM;     // 100000 (divisible by 16)
    const int E = in_sizes[2];           // 1600000
    const long long* src = ei;
    const long long* dst = ei + E;

    float* ws   = (float*)d_ws;
    float* deg  = ws;  ws += N;
    float* nrm  = ws;  ws += E;
    float* buf0 = ws;  ws += (size_t)N * DIM;
    float* buf1 = ws;  ws += (size_t)N * DIM;
    float* acc  = ws;  ws += (size_t)N * DIM;
    float* act  = ws;  ws += (size_t)N * DIM;

    const int T = 256;
    auto cdiv = [](long long a, long long b) { return (int)((a + b - 1) / b); };
    const int nTiles = N / 16;
    const int gGemm  = cdiv((long long)nTiles * 32, T);

    // GCN normalization
    zero_kernel<<<cdiv(N, T), T, 0, stream>>>(deg, N);
    deg_kernel<<<cdiv(E, T), T, 0, stream>>>(dst, ew, deg, E);
    norm_kernel<<<cdiv(E, T), T, 0, stream>>>(src, dst, ew, deg, nrm, E);

    // ---- TAGConv layer 0 ----
    tag_gemm_kernel<<<gGemm, T, 0, stream>>>(x, W0, b0, acc, nTiles, 0);
    {
        const float* hprev = x;
        float* hb[2] = {buf0, buf1};
        for (int k = 1; k <= 3; ++k) {
            float* hn = hb[(k - 1) & 1];
            zero_kernel<<<cdiv((long long)N * DIM, T), T, 0, stream>>>(hn, N * DIM);
            prop_kernel<<<cdiv((long long)E * 8, T), T, 0, stream>>>(src, dst, nrm, hprev, hn, E);
            tag_gemm_kernel<<<gGemm, T, 0, stream>>>(hn, W0 + k * DIM * DIM, b0, acc, nTiles, 1);
            hprev = hn;
        }
    }

    // BatchNorm(N) + LeakyReLU (dropout p=0 -> identity)
    bn_lrelu_kernel<<<cdiv((long long)N * DIM / DIM * 32, T), T, 0, stream>>>(acc, gamma, beta, act, N);

    // ---- TAGConv layer 1 ----
    tag_gemm_kernel<<<gGemm, T, 0, stream>>>(act, W1, b1, acc, nTiles, 0);
    {
        const float* hprev = act;
        float* hb[2] = {buf0, buf1};
        for (int k = 1; k <= 3; ++k) {
            float* hn = hb[(k - 1) & 1];
            zero_kernel<<<cdiv((long long)N * DIM, T), T, 0, stream>>>(hn, N * DIM);
            prop_kernel<<<cdiv((long long)E * 8, T), T, 0, stream>>>(src, dst, nrm, hprev, hn, E);
            tag_gemm_kernel<<<gGemm, T, 0, stream>>>(hn, W1 + k * DIM * DIM, b1, acc, nTiles, 1);
            hprev = hn;
        }
    }

    // Final feature mask
    mask_kernel<<<cdiv((long long)N * DIM, T), T, 0, stream>>>(acc, fmask, (float*)d_out, N * DIM);
}